// HungarianLoss_76433238000156
// MI455X (gfx1250) — compile-verified
//
#include <hip/hip_runtime.h>
#include <hip/hip_bf16.h>
#include <math.h>

// Problem shape (fixed by setup_inputs in the reference)
#define BB 64
#define NN 100
#define MM 100
#define CC 92
#define EPSV 1e-7f
#define HBD 128  // threads per Hungarian workgroup (4 wave32s)

typedef float v2f_t __attribute__((ext_vector_type(2)));
typedef float v8f_t __attribute__((ext_vector_type(8)));

// ---------------------------------------------------------------------------
// DIoU + smooth-L1(mean over 4 xyxy coords) pair loss; inputs are cxcywh.
// ---------------------------------------------------------------------------
__device__ __forceinline__ float pair_bbox_loss(const float* __restrict__ pb,
                                                const float* __restrict__ tb) {
  float px1 = pb[0] - 0.5f * pb[2], py1 = pb[1] - 0.5f * pb[3];
  float px2 = pb[0] + 0.5f * pb[2], py2 = pb[1] + 0.5f * pb[3];
  float tx1 = tb[0] - 0.5f * tb[2], ty1 = tb[1] - 0.5f * tb[3];
  float tx2 = tb[0] + 0.5f * tb[2], ty2 = tb[1] + 0.5f * tb[3];
  float iw = fmaxf(fminf(px2, tx2) - fmaxf(px1, tx1), 0.0f);
  float ih = fmaxf(fminf(py2, ty2) - fmaxf(py1, ty1), 0.0f);
  float inter = iw * ih;
  float uni = (px2 - px1) * (py2 - py1) + (tx2 - tx1) * (ty2 - ty1) - inter;
  float iou = inter / (uni + EPSV);
  float cw = fmaxf(px2, tx2) - fminf(px1, tx1);
  float ch = fmaxf(py2, ty2) - fminf(py1, ty1);
  float diag2 = cw * cw + ch * ch + EPSV;
  float dx = 0.5f * (px1 + px2) - 0.5f * (tx1 + tx2);
  float dy = 0.5f * (py1 + py2) - 0.5f * (ty1 + ty2);
  float diou = 1.0f - iou + (dx * dx + dy * dy) / diag2;
  float s = 0.0f;
  float dp[4] = {px1 - tx1, py1 - ty1, px2 - tx2, py2 - ty2};
#pragma unroll
  for (int k = 0; k < 4; ++k) {
    float d = dp[k], ad = fabsf(d);
    s += (ad < 1.0f) ? 0.5f * d * d : ad - 0.5f;
  }
  return diou + 0.25f * s;
}

// ---------------------------------------------------------------------------
// Kernel 1: cost[b,n,m] via WMMA one-hot GEMM (class term) + fused bbox term.
// One wave32 per 16x16 output tile; grid = (7, 7, 64).
//   A[n,c] = (targ_cat[b,n]==c) ? -1 : 0   (synthesized in registers, exact)
//   B[c,m] = pred_cat[b,m,c]               (transposed gather, b64 per step)
//   D = A*B  ->  -pred_cat[b, m, targ_cat[b,n]]
// OOB tile rows/cols produce garbage in D rows/cols that are never stored,
// so no masking of A/B values is required (only clamped addresses).
// ---------------------------------------------------------------------------
__global__ __launch_bounds__(32) void cost_tile_kernel(
    const int* __restrict__ tcat, const float* __restrict__ tbox,
    const float* __restrict__ pcat, const float* __restrict__ pbox,
    float* __restrict__ cost) {
  const int lane = threadIdx.x;  // 0..31, wave32
  const int tm = blockIdx.x;     // pred tile
  const int tn = blockIdx.y;     // target tile
  const int b  = blockIdx.z;
  const int n0 = tn * 16, m0 = tm * 16;
  const int hi = lane >> 4, lo = lane & 15;

  // A-matrix row for this lane (16x4 f32 layout: lanes 0-15/16-31 both M=lo,
  // K = v + 2*hi). Clamp index -> unconditional load; OOB rows never stored.
  const int nA  = n0 + lo;
  const int tcA = tcat[b * NN + (nA < NN ? nA : NN - 1)];
  // B-matrix column for this lane (4x16 f32 layout: N=lo, K = v + 2*hi).
  const int mB = m0 + lo;
  const float* __restrict__ pcol =
      pcat + ((size_t)b * MM + (mB < MM ? mB : MM - 1)) * CC;

  // Two independent accumulator chains to break the WMMA RAW dependency.
  v8f_t acc0 = {}, acc1 = {};
#pragma unroll
  for (int it = 0; it < CC / 4; ++it) {  // 23 K-steps of 4
    const int k0 = it * 4;
    const int kL = k0 + 2 * hi;
    // Unconditional aligned 8-byte load: floats kL, kL+1 of this pred row.
    v2f_t bv = *(const v2f_t*)(pcol + kL);
    v2f_t av;
    av[0] = (tcA == kL)     ? -1.0f : 0.0f;
    av[1] = (tcA == kL + 1) ? -1.0f : 0.0f;
    if (it & 1)
      acc1 = __builtin_amdgcn_wmma_f32_16x16x4_f32(false, av, false, bv,
                                                   (short)0, acc1, false, false);
    else
      acc0 = __builtin_amdgcn_wmma_f32_16x16x4_f32(false, av, false, bv,
                                                   (short)0, acc0, false, false);
  }
  v8f_t acc = acc0 + acc1;

  // C/D layout: VGPR r, lane -> (M = r + 8*hi, N = lo). Fuse bbox cost + store.
#pragma unroll
  for (int r = 0; r < 8; ++r) {
    int n = n0 + r + 8 * hi;
    int m = m0 + lo;
    // targ_cat for row r+8*hi lives in lane r+8*hi's tcA (lanes 0-15 = rows).
    int tc = __shfl(tcA, r + 8 * hi, 32);
    if (n < NN && m < MM) {
      float cv = acc[r];
      if (tc == 0)  // faithful to reference: bbox term masked by (targ_cat==0)
        cv += pair_bbox_loss(pbox + ((size_t)b * MM + m) * 4,
                             tbox + ((size_t)b * NN + n) * 4);
      cost[((size_t)b * NN + n) * MM + m] = cv;
    }
  }
}

// ---------------------------------------------------------------------------
// Kernel 2: exact Hungarian (shortest augmenting path), one workgroup/batch.
// Column scans and arg-min parallelized over 128 threads; state in LDS.
// Tie-breaking matches np.argmin (first minimum = smallest index).
// ---------------------------------------------------------------------------
__global__ __launch_bounds__(HBD) void hungarian_kernel(
    const float* __restrict__ cost, int* __restrict__ assign) {
  const int b = blockIdx.x;
  const int tid = threadIdx.x;
  const int n = NN;  // square N==M
  const float* __restrict__ Cb = cost + (size_t)b * NN * MM;
  const float FINF = 3.0e38f;

  __shared__ float u[NN + 1], v[NN + 1], minv[NN + 1];
  __shared__ int p[NN + 1], way[NN + 1];
  __shared__ unsigned char used[NN + 1];
  __shared__ float rv[HBD];
  __shared__ int ri[HBD];
  __shared__ int s_j0;

  for (int j = tid; j <= n; j += HBD) { u[j] = 0.f; v[j] = 0.f; p[j] = 0; }
  __syncthreads();

  for (int i = 1; i <= n; ++i) {
    if (tid == 0) { p[0] = i; s_j0 = 0; }
    for (int j = tid; j <= n; j += HBD) { minv[j] = FINF; way[j] = 0; used[j] = 0; }
    // Prefetch next outer row of the cost matrix into the WGP cache; it is
    // guaranteed to be scanned first in the next augmenting-path solve.
    if (i < n && tid < MM) __builtin_prefetch(&Cb[(size_t)i * MM + tid], 0, 3);
    __syncthreads();

    bool done = false;
    while (!done) {
      int j0 = s_j0;
      if (tid == 0) used[j0] = 1;
      __syncthreads();
      int i0 = p[j0];
      float ui0 = u[i0];

      float best = FINF; int bestj = 0;
      for (int j = tid; j <= n; j += HBD) {
        if (j >= 1 && !used[j]) {
          float cur = Cb[(size_t)(i0 - 1) * MM + (j - 1)] - ui0 - v[j];
          if (cur < minv[j]) { minv[j] = cur; way[j] = j0; }
        }
        float mv = used[j] ? FINF : minv[j];
        if (mv < best) { best = mv; bestj = j; }  // strict <: first-min per thread
      }
      rv[tid] = best; ri[tid] = bestj;
      __syncthreads();
      for (int s = HBD / 2; s > 0; s >>= 1) {
        if (tid < s) {
          float ov = rv[tid + s]; int oi = ri[tid + s];
          if (ov < rv[tid] || (ov == rv[tid] && oi < ri[tid])) {
            rv[tid] = ov; ri[tid] = oi;
          }
        }
        __syncthreads();
      }
      float delta = rv[0];
      int j1 = ri[0];

      // u[p[j]] for used j hit distinct rows -> no atomics needed.
      for (int j = tid; j <= n; j += HBD) {
        if (used[j]) { u[p[j]] += delta; v[j] -= delta; }
        else         { minv[j] -= delta; }
      }
      __syncthreads();
      if (tid == 0) s_j0 = j1;
      done = (p[j1] == 0);  // uniform: j1 and p unchanged since sync
      __syncthreads();
    }

    if (tid == 0) {  // augment along the alternating path (serial, short)
      int j0 = s_j0;
      while (j0 != 0) { int j1 = way[j0]; p[j0] = p[j1]; j0 = j1; }
    }
    __syncthreads();
  }

  for (int j = 1 + tid; j <= n; j += HBD) assign[b * NN + (p[j] - 1)] = j - 1;
}

// ---------------------------------------------------------------------------
// Kernel 3: per-slot losses, one wave32 per (b,n) row for coalesced reads.
// Lane-parallel log-softmax (shfl_xor reductions) + masked bbox pair loss.
// Also emits assign (as float) into d_out[1..].
// ---------------------------------------------------------------------------
__global__ __launch_bounds__(128) void loss_slot_kernel(
    const int* __restrict__ tcat, const float* __restrict__ tbox,
    const float* __restrict__ pcat, const float* __restrict__ pbox,
    const int* __restrict__ assign, float* __restrict__ acc,
    float* __restrict__ out_assign) {
  const int wave = (int)((blockIdx.x * blockDim.x + threadIdx.x) >> 5);
  const int lane = threadIdx.x & 31;
  if (wave >= BB * NN) return;
  const int b = wave / NN;
  const int a = assign[wave];
  const float* __restrict__ row = pcat + ((size_t)b * MM + a) * CC;

  float mx = -3.0e38f;
  for (int c = lane; c < CC; c += 32) mx = fmaxf(mx, row[c]);
#pragma unroll
  for (int o = 16; o > 0; o >>= 1) mx = fmaxf(mx, __shfl_xor(mx, o, 32));
  float s = 0.0f;
  for (int c = lane; c < CC; c += 32) s += expf(row[c] - mx);
#pragma unroll
  for (int o = 16; o > 0; o >>= 1) s += __shfl_xor(s, o, 32);

  if (lane == 0) {
    out_assign[wave] = (float)a;
    int tc = tcat[wave];
    float nll = (logf(s) + mx) - row[tc];
    atomicAdd(&acc[0], nll * (1.0f / (float)(BB * NN)));
    if (tc != 0) {
      float lb = pair_bbox_loss(pbox + ((size_t)b * MM + a) * 4,
                                tbox + (size_t)wave * 4);
      atomicAdd(&acc[1], lb);
      atomicAdd((unsigned int*)(acc + 2), 1u);
    }
  }
}

__global__ void finalize_kernel(const float* __restrict__ acc,
                                float* __restrict__ out) {
  if (threadIdx.x == 0 && blockIdx.x == 0) {
    unsigned cnt = ((const unsigned int*)acc)[2];
    out[0] = acc[0] + acc[1] / (float)cnt;  // BBOX_WEIGHT = 1
  }
}

// ---------------------------------------------------------------------------
extern "C" void kernel_launch(void* const* d_in, const int* in_sizes, int n_in,
                              void* d_out, int out_size, void* d_ws,
                              size_t ws_size, hipStream_t stream) {
  (void)in_sizes; (void)n_in; (void)out_size; (void)ws_size;
  const int*   tcat = (const int*)d_in[0];
  const float* tbox = (const float*)d_in[1];
  const float* pcat = (const float*)d_in[2];
  const float* pbox = (const float*)d_in[3];
  float* out = (float*)d_out;

  // Workspace layout: cost (2.56 MB, L2-resident) | assign | accumulators
  char* ws = (char*)d_ws;
  float* cost = (float*)ws;
  size_t costBytes = (size_t)BB * NN * MM * sizeof(float);
  int* assign = (int*)(ws + costBytes);
  size_t assignBytes = (size_t)BB * NN * sizeof(int);
  float* acc = (float*)(ws + costBytes + assignBytes);

  hipMemsetAsync(acc, 0, 16, stream);  // graph-capturable

  dim3 gridc((MM + 15) / 16, (NN + 15) / 16, BB);
  cost_tile_kernel<<<gridc, 32, 0, stream>>>(tcat, tbox, pcat, pbox, cost);
  hungarian_kernel<<<BB, HBD, 0, stream>>>(cost, assign);
  int waves = BB * NN;  // one wave32 per slot
  loss_slot_kernel<<<(waves * 32 + 127) / 128, 128, 0, stream>>>(
      tcat, tbox, pcat, pbox, assign, acc, out + 1);
  finalize_kernel<<<1, 1, 0, stream>>>(acc, out);
}